// NGCF_27608049779427
// MI455X (gfx1250) — compile-verified
//
#include <hip/hip_runtime.h>
#include <hip/hip_bf16.h>

typedef float v2f __attribute__((ext_vector_type(2)));
typedef float v8f __attribute__((ext_vector_type(8)));

#define NU 50000
#define NI 100000
#define NT 150000          // NU + NI
#define DD 64
#define NNZE 2400000
#define BATCH 4096
#define SLOPE 0.01f

// ---------------------------------------------------------------------------
// Layer-0 gather: out cols [0,64) come straight from the input embeddings
// (reference appends the *un-normalized* initial ego).
// ---------------------------------------------------------------------------
__global__ __launch_bounds__(256)
void k_gather0(const float* __restrict__ ue, const float* __restrict__ ie,
               const int* __restrict__ us, const int* __restrict__ pi,
               const int* __restrict__ ni, float* __restrict__ out) {
    int t = blockIdx.x * 256 + threadIdx.x;        // 3*4096*64 threads
    if (t >= 3 * BATCH * DD) return;
    int c  = t & 63;
    int b  = t >> 6;                               // 0..12287 = seg*4096+bb
    int seg = b >> 12;
    int bb  = b & (BATCH - 1);
    float v;
    if (seg == 0)      v = ue[(size_t)us[bb] * DD + c];
    else if (seg == 1) v = ie[(size_t)pi[bb] * DD + c];
    else               v = ie[(size_t)ni[bb] * DD + c];
    out[(size_t)b * 256 + c] = v;
}

// ---------------------------------------------------------------------------
// SpMM scatter: L[rows[e]] += vals[e] * ego[cols[e]].  16 threads/edge,
// float4 gather (coalesced 256B per edge), f32 global atomics into L2-resident
// L_side (ego+L = 77MB < 192MB L2).
// ---------------------------------------------------------------------------
__global__ __launch_bounds__(256)
void k_spmm(const int* __restrict__ rows, const int* __restrict__ cols,
            const float* __restrict__ vals, const float* __restrict__ ego,
            float* __restrict__ L) {
    long t = (long)blockIdx.x * 256 + threadIdx.x; // NNZE*16 work items
    if (t >= (long)NNZE * 16) return;
    int e = (int)(t >> 4);
    int q = ((int)t & 15) * 4;
    int r = rows[e], c = cols[e];
    float v = vals[e];
    const float4 x = *(const float4*)(ego + (size_t)c * DD + q);
    float* dst = L + (size_t)r * DD + q;
    atomicAdd(dst + 0, v * x.x);
    atomicAdd(dst + 1, v * x.y);
    atomicAdd(dst + 2, v * x.z);
    atomicAdd(dst + 3, v * x.w);
}

// ---------------------------------------------------------------------------
// Dense layer, fp32 WMMA (V_WMMA_F32_16X16X4_F32):
//   ego[i] = leaky_relu( (L[i]+ego[i]) @ Wgc + (L[i]*ego[i]) @ Wbi + 2*b )
// One wave per 16-row M-tile; 4 N-tiles of 16; K-loop of 16 (K=4 per WMMA).
// Both matmuls accumulate into the same v8f accumulators.
// Weights swizzled in LDS so a B fragment (K-pair x 16 cols) is one b64 load.
// ---------------------------------------------------------------------------
__global__ __launch_bounds__(256)
void k_dense_wmma(const float* __restrict__ L, float* __restrict__ ego,
                  const float* __restrict__ Wgc, const float* __restrict__ Wbi,
                  const float* __restrict__ bias) {
    __shared__ float sg[4096];   // swizzled W_gc: [(r>>1)*64 + c]*2 + (r&1)
    __shared__ float sb[4096];   // swizzled W_bi
    __shared__ float sbias[64];  // 2*b_bi (bias added to both branches)

    for (int i = threadIdx.x; i < 4096; i += 256) {
        int r = i >> 6, c = i & 63;
        int dst = ((((r >> 1) * 64) + c) << 1) | (r & 1);
        sg[dst] = Wgc[i];
        sb[dst] = Wbi[i];
    }
    if (threadIdx.x < 64) sbias[threadIdx.x] = 2.0f * bias[threadIdx.x];
    __syncthreads();

    const int wave  = threadIdx.x >> 5;
    const int lane  = threadIdx.x & 31;
    const int mbase = (blockIdx.x * 8 + wave) * 16;
    if (mbase >= NT) return;                 // wave-uniform: EXEC stays all-1s

    const int half = lane >> 4;              // selects K pair (A) / row pair (D)
    const int n    = lane & 15;

    const float* lrow = L   + (size_t)(mbase + n) * DD;  // A: lane holds row M=n
    const float* erow = ego + (size_t)(mbase + n) * DD;

    v8f acc[4];
    #pragma unroll
    for (int j = 0; j < 4; ++j) acc[j] = (v8f){};

    #pragma unroll
    for (int kk = 0; kk < 16; ++kk) {
        int off = kk * 4 + half * 2;         // A frag: K = off, off+1
        v2f ls = *(const v2f*)(lrow + off);
        v2f eg = *(const v2f*)(erow + off);
        v2f a_li  = ls + eg;                 // (L + I) @ E branch
        v2f a_mul = ls * eg;                 // bilinear branch
        int bbase = ((kk * 2 + half) * 64) << 1;  // float index into swizzle
        #pragma unroll
        for (int j = 0; j < 4; ++j) {
            v2f bgc = *(const v2f*)(sg + bbase + ((j * 16 + n) << 1));
            v2f bbi = *(const v2f*)(sb + bbase + ((j * 16 + n) << 1));
            acc[j] = __builtin_amdgcn_wmma_f32_16x16x4_f32(
                false, a_li, false, bgc, (short)0, acc[j], false, false);
            acc[j] = __builtin_amdgcn_wmma_f32_16x16x4_f32(
                false, a_mul, false, bbi, (short)0, acc[j], false, false);
        }
    }

    // Epilogue: +2*b, leaky_relu, in-place store.
    // D layout: VGPR v -> row mbase + v + 8*half, col j*16 + n.
    #pragma unroll
    for (int j = 0; j < 4; ++j) {
        float bsum = sbias[j * 16 + n];
        #pragma unroll
        for (int v = 0; v < 8; ++v) {
            float x = acc[j][v] + bsum;
            x = (x >= 0.f) ? x : SLOPE * x;
            ego[(size_t)(mbase + (half << 3) + v) * DD + j * 16 + n] = x;
        }
    }
}

// ---------------------------------------------------------------------------
// Gather + row-normalize: norms are only needed for the 3*4096 gathered rows.
// One wave per output row; 32 lanes x float2; shfl-xor reduction (wave32).
// ---------------------------------------------------------------------------
__global__ __launch_bounds__(256)
void k_gather_norm(const float* __restrict__ ego, const int* __restrict__ us,
                   const int* __restrict__ pi, const int* __restrict__ ni,
                   float* __restrict__ out, int colbase) {
    int wid  = (blockIdx.x * 256 + threadIdx.x) >> 5;   // 0..12287
    int lane = threadIdx.x & 31;
    if (wid >= 3 * BATCH) return;
    int seg = wid >> 12;
    int bb  = wid & (BATCH - 1);
    int node;
    if (seg == 0)      node = us[bb];
    else if (seg == 1) node = NU + pi[bb];
    else               node = NU + ni[bb];
    v2f x = *(const v2f*)(ego + (size_t)node * DD + lane * 2);
    float ss = x.x * x.x + x.y * x.y;
    #pragma unroll
    for (int o = 16; o > 0; o >>= 1) ss += __shfl_xor(ss, o, 32);
    float scale = 1.0f / fmaxf(sqrtf(ss), 1e-12f);
    v2f y; y.x = x.x * scale; y.y = x.y * scale;
    *(v2f*)(out + (size_t)wid * 256 + colbase + lane * 2) = y;
}

// ---------------------------------------------------------------------------
extern "C" void kernel_launch(void* const* d_in, const int* in_sizes, int n_in,
                              void* d_out, int out_size, void* d_ws, size_t ws_size,
                              hipStream_t stream) {
    const float* ue   = (const float*)d_in[0];   // [50000,64]
    const float* ie   = (const float*)d_in[1];   // [100000,64]
    const float* Wgc  = (const float*)d_in[2];   // [3,64,64]
    const float* Wbi  = (const float*)d_in[3];   // [3,64,64]
    const float* bbi  = (const float*)d_in[4];   // [3,1,64]
    const float* vals = (const float*)d_in[5];   // [NNZ]
    const int*   rows = (const int*)d_in[6];
    const int*   cols = (const int*)d_in[7];
    const int*   us   = (const int*)d_in[8];
    const int*   pi   = (const int*)d_in[9];
    const int*   ni   = (const int*)d_in[10];
    float* out = (float*)d_out;

    float* ego = (float*)d_ws;                    // [150000,64] = 38.4 MB
    float* L   = ego + (size_t)NT * DD;           // [150000,64] = 38.4 MB

    // ego = concat(user_emb, item_emb)
    hipMemcpyAsync(ego, ue, (size_t)NU * DD * sizeof(float),
                   hipMemcpyDeviceToDevice, stream);
    hipMemcpyAsync(ego + (size_t)NU * DD, ie, (size_t)NI * DD * sizeof(float),
                   hipMemcpyDeviceToDevice, stream);

    // layer-0 columns of the output
    k_gather0<<<(3 * BATCH * DD + 255) / 256, 256, 0, stream>>>(
        ue, ie, us, pi, ni, out);

    for (int k = 0; k < 3; ++k) {
        hipMemsetAsync(L, 0, (size_t)NT * DD * sizeof(float), stream);
        long tot = (long)NNZE * 16;
        k_spmm<<<(int)((tot + 255) / 256), 256, 0, stream>>>(
            rows, cols, vals, ego, L);
        k_dense_wmma<<<(NT + 127) / 128, 256, 0, stream>>>(
            L, ego, Wgc + k * 4096, Wbi + k * 4096, bbi + k * 64);
        k_gather_norm<<<(3 * BATCH * 32 + 255) / 256, 256, 0, stream>>>(
            ego, us, pi, ni, out, 64 * (k + 1));
    }
}